// RoiPoolingConv_37503654428808
// MI455X (gfx1250) — compile-verified
//
#include <hip/hip_runtime.h>
#include <cfloat>

// Problem constants (from the reference)
#define R_NUM 128
#define C_NUM 256
#define H_NUM 64
#define W_NUM 64
#define P_NUM 7

// One block per (roi, channel). 64 threads = 2 wave32 waves.
// Phase 1: async-stage only the ROI's bounding-box rows/column-chunks into LDS
//          with GLOBAL_LOAD_ASYNC_TO_LDS_B128 (CDNA5 async path, ASYNCcnt).
// Phase 2: 49 threads compute the 7x7 bin maxima from LDS.
__global__ __launch_bounds__(64) void roi_maxpool_gfx1250(
    const float* __restrict__ x,    // [256, 64, 64]
    const int*   __restrict__ rois, // [128, 4] = (x, y, w, h)
    float*       __restrict__ out)  // [128, 256, 7, 7]
{
    __shared__ float tile[H_NUM * W_NUM];   // 16 KB: up to 64 staged rows, stride 64

    const int bid = blockIdx.x;
    const int c   = bid & (C_NUM - 1);
    const int r   = bid >> 8;
    const int t   = (int)threadIdx.x;

    const int rx = rois[r * 4 + 0];
    const int ry = rois[r * 4 + 1];
    const int rw = rois[r * 4 + 2];
    const int rh = rois[r * 4 + 3];

    // rl = w / 7 in f32, exactly as the reference (IEEE f32 divide).
    const float rl_x = (float)rw / 7.0f;
    const float rl_y = (float)rh / 7.0f;
    const int   flx  = (int)floorf(rl_x);   // bin extent (cols), 1..8
    const int   fly  = (int)floorf(rl_y);   // bin extent (rows), 1..8

    // Rows needed: [ry, ye); cols needed: [rx, xe).
    // Use un-fused mul+add to match the reference's separate mul/add.
    const int ylast = (int)floorf(__fadd_rn((float)ry, __fmul_rn(6.0f, rl_y)));
    const int ye    = min(ylast + fly, H_NUM);
    const int nrows = ye - ry;              // 1..64

    const int xlast = (int)floorf(__fadd_rn((float)rx, __fmul_rn(6.0f, rl_x)));
    const int xe    = min(xlast + flx, W_NUM);
    // 16-byte (4-float) chunks actually covering [rx, xe):
    const int c0 = rx >> 2;                 // first chunk index (aligned down)
    const int c1 = (xe + 3) >> 2;           // one past last chunk index

    // ---------- Phase 1: async global -> LDS staging ----------
    // Low 32 bits of a generic LDS pointer == LDS byte offset (flat aperture).
    const unsigned ldsBase = (unsigned)(unsigned long long)(&tile[0]);
    const float*   gbase   = x + ((size_t)c << 12) + (size_t)(ry * W_NUM);

    const int rowBlk = t >> 4;   // 0..3 : row within a 4-row slab
    const int col16  = t & 15;   // 0..15: 16-byte column chunk within a row
    const bool colActive = (col16 >= c0) && (col16 < c1);

    for (int it = 0; it * 4 < nrows; ++it) {
        const int row = it * 4 + rowBlk;
        if ((row < nrows) && colActive) {
            const unsigned off = (unsigned)(row * (W_NUM * 4) + col16 * 16); // bytes
            const unsigned la  = ldsBase + off;
            // GVS mode: 64-bit SGPR base + 32-bit VGPR byte offset; VDST = LDS addr.
            asm volatile("global_load_async_to_lds_b128 %0, %1, %2"
                         :
                         : "v"(la), "v"(off), "s"(gbase)
                         : "memory");
        }
    }
    // Wave-local completion of all async LDS writes, then WG visibility.
    asm volatile("s_wait_asynccnt 0" ::: "memory");
    __syncthreads();

    // ---------- Phase 2: per-bin max over LDS ----------
    if (t < P_NUM * P_NUM) {
        const int py = t / P_NUM;
        const int px = t % P_NUM;

        // starts = floor(coord + i*rl); ends = starts + floor(rl);
        // starts = max(starts,0); ends = min(ends, limit)   (reference order)
        int x1 = (int)floorf(__fadd_rn((float)rx, __fmul_rn((float)px, rl_x)));
        int x2 = x1 + flx;
        x1 = max(x1, 0);
        x2 = min(x2, W_NUM);

        int y1 = (int)floorf(__fadd_rn((float)ry, __fmul_rn((float)py, rl_y)));
        int y2 = y1 + fly;
        y1 = max(y1, 0);
        y2 = min(y2, H_NUM);

        float m = -FLT_MAX;   // matches jnp.finfo(float32).min
        for (int iy = y1; iy < y2; ++iy) {
            const float* rowp = &tile[(iy - ry) * W_NUM];
            for (int ix = x1; ix < x2; ++ix) {
                m = fmaxf(m, rowp[ix]);
            }
        }

        // out[1, R, C, 7, 7] flat: ((r*C + c)*49 + py*7 + px)
        out[((size_t)(r * C_NUM + c)) * (P_NUM * P_NUM) + (size_t)t] = m;
    }
}

extern "C" void kernel_launch(void* const* d_in, const int* in_sizes, int n_in,
                              void* d_out, int out_size, void* d_ws, size_t ws_size,
                              hipStream_t stream) {
    (void)in_sizes; (void)n_in; (void)out_size; (void)d_ws; (void)ws_size;

    const float* x    = (const float*)d_in[0];   // [1,256,64,64] f32
    const int*   rois = (const int*)d_in[1];     // [128,4] i32
    float*       out  = (float*)d_out;           // [1,128,256,7,7] f32

    dim3 grid(R_NUM * C_NUM);   // 32768 blocks: one per (roi, channel)
    dim3 block(64);             // 2 wave32 waves
    hipLaunchKernelGGL(roi_maxpool_gfx1250, grid, block, 0, stream, x, rois, out);
}